// CIN_3212635538196
// MI455X (gfx1250) — compile-verified
//
#include <hip/hip_runtime.h>

// ---------------------------------------------------------------------------
// Fused CIN forward for MI455X (gfx1250, wave32, WMMA).
//
// cur_l[b,n,d] = relu( sum_{h,m} W_l[h*M+m,n] * x0[b,h,d] * prev[b,m,d] )
// out[b, 0:128]   = sum_d cur_0[b,:,d]
// out[b, 128:256] = sum_d cur_1[b,:,d]
//
// ~172 GFLOP f32 vs ~23 MB of mandatory HBM traffic -> compute bound.
// Per d-slice: GEMM C[16b x 128n] += A[16b x K] * W[K x 128n], K = F*M, with
// rank-1 structured A[b,(h,m)] = x0[b,h,d]*prev[b,m,d]:
//   * A-fragments for V_WMMA_F32_16X16X4_F32 are built in-register as
//     (per-lane x0 scalar) * (register-resident prev fragment) -> z (2 GB for
//     layer 1) is never materialized.
//   * Both layers fused: layer-0 output crosses to layer 1 through a 64 KB
//     LDS transpose buffer only (cur0 never touches HBM).
//   * W_h tiles are double-buffered in LDS and staged with
//     GLOBAL_LOAD_ASYNC_TO_LDS_B128 (ASYNCcnt) when available, so the 64 KB/h
//     staging latency hides under the 256 WMMAs of the previous tile.
//   * d-sums are written as per-d-group partials and reduced by a second
//     kernel -> bitwise deterministic across graph replays (no FP atomics).
// ---------------------------------------------------------------------------

typedef float v2f __attribute__((ext_vector_type(2)));
typedef float v8f __attribute__((ext_vector_type(8)));
typedef int v4i __attribute__((vector_size(16)));

#define B_TOT 4096
#define F_IN  32
#define DDIM  32
#define NOUT  128

#if __has_builtin(__builtin_amdgcn_global_load_async_to_lds_b128)
#define ASYNC_LDS 1
typedef __attribute__((address_space(1))) v4i gas_v4i;  // global (device) AS
typedef __attribute__((address_space(3))) v4i las_v4i;  // LDS (shared) AS
#else
#define ASYNC_LDS 0
#endif

template <int N>
__device__ __forceinline__ void async_wait() {
#if ASYNC_LDS
#if __has_builtin(__builtin_amdgcn_s_wait_asynccnt)
  __builtin_amdgcn_s_wait_asynccnt(N);
#else
  asm volatile("s_wait_asynccnt %0" ::"i"(N) : "memory");
#endif
#endif
}

// Stage one W_h tile (NCHUNK float4 per thread, 256 threads) into LDS.
template <int NCHUNK>
__device__ __forceinline__ void stage_issue(const float* __restrict__ gsrc,
                                            float* ldst, int tid) {
#if ASYNC_LDS
#pragma unroll
  for (int t = 0; t < NCHUNK; ++t) {
    const int c = (tid + t * 256) * 4;
    __builtin_amdgcn_global_load_async_to_lds_b128(
        (gas_v4i*)(v4i*)(void*)(gsrc + c), (las_v4i*)(v4i*)(void*)(ldst + c),
        0, 0);
  }
#else
  const float4* src = (const float4*)gsrc;
  float4* dst = (float4*)ldst;
#pragma unroll
  for (int t = 0; t < NCHUNK; ++t) dst[tid + t * 256] = src[tid + t * 256];
#endif
}

// One CIN layer: acc[16b x 128n] += sum_{h,m} W[h*M+m,n]*x0[b,h,d]*prev[b,m,d]
// W_h tiles double-buffered between buf0/buf1; prev fragment lives in VGPRs.
template <int M>
__device__ __forceinline__ void layer_accum(
    const float* __restrict__ X, const float* __restrict__ W, float* buf0,
    float* buf1, const float (&prevReg)[M / 2], v8f (&acc)[8], int b, int d,
    int tid, int ln, int krow) {
  constexpr int NCHUNK = (M * NOUT) / (4 * 256);  // float4 per thread per tile

  stage_issue<NCHUNK>(W, buf0, tid);  // prologue: tile h=0
  for (int h = 0; h < F_IN; ++h) {
    float* cur = (h & 1) ? buf1 : buf0;
    float* nxt = (h & 1) ? buf0 : buf1;
    if (h + 1 < F_IN) {
      stage_issue<NCHUNK>(W + (size_t)(h + 1) * M * NOUT, nxt, tid);
      async_wait<NCHUNK>();  // tile h landed; tile h+1 still in flight
    } else {
      async_wait<0>();
    }
    __syncthreads();  // all waves' tile-h copies visible

    const float s = X[(b * F_IN + h) * DDIM + d];  // x0[b,h,d] per-lane scalar
#pragma unroll
    for (int i = 0; i < M / 4; ++i) {  // K step of 4 (m0 = 4*i)
      v2f a;                           // rank-1 A fragment: s * prev
      a.x = s * prevReg[2 * i + 0];
      a.y = s * prevReg[2 * i + 1];
      const float* wrow = cur + (i * 4) * NOUT + krow + ln;
#pragma unroll
      for (int nt = 0; nt < 8; ++nt) {  // 8 n-tiles of 16
        v2f bf;
        bf.x = wrow[nt * 16];
        bf.y = wrow[nt * 16 + NOUT];
        acc[nt] = __builtin_amdgcn_wmma_f32_16x16x4_f32(
            false, a, false, bf, (short)0, acc[nt], false, false);
      }
    }
    __syncthreads();  // compute(h) done before tile h+2 overwrites `cur`
  }
}

// relu -> curbuf[d_local][m(128)][b(16)] transpose buffer -> per-d-group
// partial d-sums for output columns [col0, col0+128).
__device__ __forceinline__ void emit_partials(v8f (&acc)[8], float* curbuf,
                                              float* part, int tid, int wave,
                                              int ln, int brow, int b0, int g,
                                              int col0) {
#pragma unroll
  for (int nt = 0; nt < 8; ++nt) {
#pragma unroll
    for (int j = 0; j < 8; ++j) {
      float v = acc[nt][j];
      v = v > 0.f ? v : 0.f;
      curbuf[wave * 2048 + (nt * 16 + ln) * 16 + (j + brow)] = v;
    }
  }
  __syncthreads();
#pragma unroll
  for (int p = 0; p < 8; ++p) {
    const int idx = tid + p * 256;  // (b:16, n:128) pairs
    const int n = idx & 127;
    const int bb = idx >> 7;
    float ssum = 0.f;
#pragma unroll
    for (int w = 0; w < 8; ++w) ssum += curbuf[w * 2048 + n * 16 + bb];
    part[((size_t)g * B_TOT + (b0 + bb)) * 256 + col0 + n] = ssum;
  }
}

__global__ __launch_bounds__(256) void cin_fused_kernel(
    const float* __restrict__ X,   // [4096, 32, 32]
    const float* __restrict__ W0,  // [1024, 128]
    const float* __restrict__ W1,  // [4096, 128]
    float* __restrict__ part)      // [4][4096][256] partial d-sums
{
  extern __shared__ __align__(16) float lds[];
  float* curbuf = lds + 16384;  // 64 KB: [d_local][m][b]; also W buf1 phase B

  const int tid = threadIdx.x;
  const int wave = tid >> 5;
  const int lane = tid & 31;
  const int ln = lane & 15;              // column / batch-row selector
  const int koff = (lane < 16) ? 0 : 2;  // K sub-row pair of this half-wave
  const int brow = (lane < 16) ? 0 : 8;  // C-layout row offset
  const int b0 = blockIdx.x * 16;        // batch tile
  const int g = blockIdx.y;              // d-group (4 groups of 8)
  const int d = g * 8 + wave;            // this wave's depth slice
  const int b = b0 + ln;                 // this lane's batch row
  const int krow = koff * NOUT;

  const v8f vzero = {0.f, 0.f, 0.f, 0.f, 0.f, 0.f, 0.f, 0.f};
  v8f acc[8];

  // ====================== Layer 0 (M = 32, prev = x0) ======================
  float prevA[16];
#pragma unroll
  for (int i = 0; i < 8; ++i) {
    const int m = i * 4 + koff;
    prevA[2 * i + 0] = X[(b * F_IN + m) * DDIM + d];
    prevA[2 * i + 1] = X[(b * F_IN + m + 1) * DDIM + d];
  }
#pragma unroll
  for (int nt = 0; nt < 8; ++nt) acc[nt] = vzero;

  // phase A double buffer: two 16 KB tiles inside the first 64 KB region
  layer_accum<32>(X, W0, lds, lds + 32 * NOUT, prevA, acc, b, d, tid, ln, krow);
  emit_partials(acc, curbuf, part, tid, wave, ln, brow, b0, g, 0);

  // =================== Layer 1 (M = 128, prev = relu(cur0)) ================
  float prevB[64];
#pragma unroll
  for (int i = 0; i < 32; ++i) {
    const int m = i * 4 + koff;
    prevB[2 * i + 0] = curbuf[wave * 2048 + (m)*16 + ln];
    prevB[2 * i + 1] = curbuf[wave * 2048 + (m + 1) * 16 + ln];
  }
#pragma unroll
  for (int nt = 0; nt < 8; ++nt) acc[nt] = vzero;
  __syncthreads();  // all prevB/partial reads of curbuf done before re-staging

  // phase B double buffer: 64 KB W region + curbuf region (prevB is in VGPRs)
  layer_accum<128>(X, W1, lds, curbuf, prevB, acc, b, d, tid, ln, krow);
  emit_partials(acc, curbuf, part, tid, wave, ln, brow, b0, g, 128);
}

// Deterministic cross-d-group reduction: out[b,n] = sum_g part[g][b][n]
__global__ __launch_bounds__(256) void cin_reduce_kernel(
    const float* __restrict__ part, float* __restrict__ out) {
  const int n = threadIdx.x;  // 0..255
  const int b = blockIdx.x;   // 0..4095
  float s = 0.f;
#pragma unroll
  for (int g = 0; g < 4; ++g) s += part[((size_t)g * B_TOT + b) * 256 + n];
  out[(size_t)b * 256 + n] = s;
}

extern "C" void kernel_launch(void* const* d_in, const int* in_sizes, int n_in,
                              void* d_out, int out_size, void* d_ws,
                              size_t ws_size, hipStream_t stream) {
  const float* X = (const float*)d_in[0];   // [4096,32,32]
  const float* W0 = (const float*)d_in[1];  // [1,1024,128]
  const float* W1 = (const float*)d_in[2];  // [1,4096,128]
  float* part = (float*)d_ws;               // 4*4096*256 f32 = 16 MB scratch
  float* out = (float*)d_out;               // [4096,256]

  dim3 grid(B_TOT / 16, 4);  // 256 batch tiles x 4 d-groups
  cin_fused_kernel<<<grid, 256, 128 * 1024, stream>>>(X, W0, W1, part);
  cin_reduce_kernel<<<B_TOT, 256, 0, stream>>>(part, out);
}